// LinearCaps2d_6081673691725
// MI455X (gfx1250) — compile-verified
//
#include <hip/hip_runtime.h>

// -----------------------------------------------------------------------------
// LinearCaps2d dynamic routing on MI455X (gfx1250, wave32).
//
// votes(16o x 16b) = W(16o x 16i) x U(16i x 16b) per (capsule, c-type), via
// V_WMMA_F32_16X16X4_F32 (4 k-steps), recomputed every routing pass so the
// 335MB votes tensor never exists; the 55MB of operands live in the 192MB L2.
// Logits are linear in v, so only Vsum[b,c,o] is carried between passes.
//
// Operand staging uses the CDNA5 Tensor Data Mover (TDM): per capsule, two
// wave-uniform tensor_load_to_lds calls DMA the contiguous 10KB weight block and
// the 16x16 strided pose tile into LDS, double-buffered and synchronized with
// s_wait_tensorcnt. TDM pad_enable inserts 1 dword after every 16, giving LDS
// rows a stride of 17 floats -> bank-conflict-free fragment reads (17 coprime
// to the 64 LDS banks) and no L2 sector amplification on the weight stream.
// -----------------------------------------------------------------------------

#define BATCH    256
#define B_TYPES  32
#define HW       64                  // H*W = 8*8
#define NCAPS    (B_TYPES * HW)      // 2048 input capsules
#define C_TYPES  10
#define POSE     16
#define NB_TILE  16                  // batch tile = WMMA N
#define N_BT     (BATCH / NB_TILE)   // 16 batch tiles
#define NCHUNK   32                  // capsule chunks per batch tile
#define CAPS_PER_CHUNK (NCAPS / NCHUNK)          // 64
#define WAVES    4
#define CAPS_PER_WAVE  (CAPS_PER_CHUNK / WAVES)  // 16
#define CAPS_EPS 1e-8f

#define WROWS    (C_TYPES * POSE)    // 160 weight rows of 16 floats per capsule
#define LROW     17                  // padded LDS row stride (floats)
#define WPAD     (WROWS * LROW)      // 2720 floats: padded weight block
#define UPAD     (NB_TILE * LROW)    // 272 floats: padded pose tile
#define SLOT     (WPAD + UPAD)       // 2992 floats per buffer slot
#define SVEC     (C_TYPES * NB_TILE * POSE)      // 2560 floats of partial s

typedef float v2f __attribute__((ext_vector_type(2)));
typedef float v8f __attribute__((ext_vector_type(8)));
typedef unsigned int v4u __attribute__((ext_vector_type(4)));
typedef int v4i __attribute__((ext_vector_type(4)));
typedef int v8i __attribute__((ext_vector_type(8)));

// ---------------------------------------------------------------------------
__global__ __launch_bounds__(256) void caps_zero_kernel(float* p, int n) {
    int g = blockIdx.x * 256 + threadIdx.x;
    if (g < n) p[g] = 0.0f;
}

// ---------------------------------------------------------------------------
// TDM descriptor group1 (see CDNA5 ISA):
//   data_size=2 (4B), pad_enable=1, pad_interval=3 (every 16 dwords),
//   pad_amount=0 (1 dword).
#define TDM_G1_DW0 ((int)((2u << 16) | (1u << 20) | (3u << 22)))

__device__ __forceinline__ void tdm_load(unsigned int lds_byte,
                                         const void* gptr, v8i g1) {
    unsigned long long ga = (unsigned long long)(uintptr_t)gptr;
    v4u g0;
    g0.x = 1u;                                    // count=1, user descriptor
    g0.y = lds_byte;                              // lds_addr (bytes)
    g0.z = (unsigned int)ga;                      // global_addr[31:0]
    g0.w = (unsigned int)(ga >> 32) | (2u << 30); // global_addr[56:32], type=2
    v4i z4 = {0, 0, 0, 0};
    v8i z8 = {0, 0, 0, 0, 0, 0, 0, 0};
    // 6-arg form on this toolchain: (g0, g1, g2, g3, g4, cpol)
    __builtin_amdgcn_tensor_load_to_lds(g0, g1, z4, z4, z8, 0);
}

// ---------------------------------------------------------------------------
// One routing pass: TDM-stage operands, recompute votes with WMMA, form
// logits from Vsum, softmax over c, accumulate partial s per (chunk, tile).
//   poses  : (256, 2048, 16)        fp32
//   weight : (2048, 10, 16, 16)     fp32  [cap][c][o][i], i contiguous
//   Vsum   : (256, 10, 16)          fp32
//   partial: (32ch, 16bt, 10, 16n, 16o)
// ---------------------------------------------------------------------------
__global__ __launch_bounds__(128) void caps_pass_kernel(
    const float* __restrict__ poses,
    const float* __restrict__ weight,
    const float* __restrict__ Vsum,
    float* __restrict__ partial)
{
    __shared__ float Vs[SVEC];                 // Vsum tile [c][n][o], 10KB
    __shared__ float pool[WAVES * 2 * SLOT];   // TDM buffers; reused as stage

    const int bt   = blockIdx.x;       // batch tile
    const int ch   = blockIdx.y;       // capsule chunk
    const int tid  = threadIdx.x;
    const int lane = tid & 31;
    const int wave = tid >> 5;
    const int half = lane >> 4;        // 0: lanes 0-15, 1: lanes 16-31
    const int n    = lane & 15;        // WMMA column (batch-in-tile) & A row

    // Preload Vsum tile for this batch tile into LDS as [c][n][o].
    for (int idx = tid; idx < SVEC; idx += 128) {
        int c  = idx >> 8;
        int r  = idx & 255;
        int nn = r >> 4;
        int oo = r & 15;
        Vs[idx] = Vsum[(bt * NB_TILE + nn) * (C_TYPES * POSE) + c * POSE + oo];
    }
    __syncthreads();

    v8f acc[C_TYPES];
#pragma unroll
    for (int c = 0; c < C_TYPES; ++c) {
        v8f z = {};
        acc[c] = z;
    }

    const int   cap0  = ch * CAPS_PER_CHUNK + wave * CAPS_PER_WAVE;
    float*      wbuf  = &pool[wave * 2 * SLOT];

    // TDM group1 descriptors (wave-uniform).
    const v8i g1_w = {TDM_G1_DW0, (int)(2560u << 16), 0, (int)(2560u << 16),
                      0, 0, 0, 0};                       // 1D: 2560 dwords
    const v8i g1_u = {TDM_G1_DW0, (int)(16u << 16), (int)(16u << 16),
                      (int)(16u << 16), 16, NCAPS * POSE, 0, 0};
                      // 2D: 16x16, row stride = 2048*16 elements

    // Issue both DMAs for one capsule slot.
    auto prefetch = [&](int jj, int slot) {
        const int cap = cap0 + jj;
        unsigned int lb = (unsigned int)(uintptr_t)&wbuf[slot * SLOT];
        tdm_load(lb, weight + (size_t)cap * (C_TYPES * POSE * POSE), g1_w);
        tdm_load(lb + WPAD * 4,
                 poses + ((size_t)(bt * NB_TILE) * NCAPS + cap) * POSE, g1_u);
    };

    auto body = [&](int j) {
        const float* wb = &wbuf[(j & 1) * SLOT];
        const float* ub = wb + WPAD;

        // B operand: lane layout N = lane%16, K = 4k + 2*half + {0,1};
        // padded rows (stride 17) -> conflict-free ds_load_b64.
        const float* up = ub + n * LROW + 2 * half;
        v2f b0 = *(const v2f*)(up + 0);
        v2f b1 = *(const v2f*)(up + 4);
        v2f b2 = *(const v2f*)(up + 8);
        v2f b3 = *(const v2f*)(up + 12);

        v8f   votes[C_TYPES];
        float logit[C_TYPES];

#pragma unroll
        for (int c = 0; c < C_TYPES; ++c) {
            const float* wc = wb + (c * POSE + n) * LROW + 2 * half;
            v2f a0 = *(const v2f*)(wc + 0);
            v2f a1 = *(const v2f*)(wc + 4);
            v2f a2 = *(const v2f*)(wc + 8);
            v2f a3 = *(const v2f*)(wc + 12);

            v8f d = {};
            d = __builtin_amdgcn_wmma_f32_16x16x4_f32(false, a0, false, b0, (short)0, d, false, false);
            d = __builtin_amdgcn_wmma_f32_16x16x4_f32(false, a1, false, b1, (short)0, d, false, false);
            d = __builtin_amdgcn_wmma_f32_16x16x4_f32(false, a2, false, b2, (short)0, d, false, false);
            d = __builtin_amdgcn_wmma_f32_16x16x4_f32(false, a3, false, b3, (short)0, d, false, false);
            votes[c] = d;

            // logit = votes(:,n) . Vsum[b,c,:]  (lane holds o = q + 8*half)
            const float* vsl = &Vs[c * (NB_TILE * POSE) + n * POSE + 8 * half];
            float dot = 0.0f;
#pragma unroll
            for (int q = 0; q < 8; ++q) dot += d[q] * vsl[q];
            dot += __shfl_xor(dot, 16, 32);
            logit[c] = dot;
        }

        // softmax over c (both o-halves compute identical weights)
        float m = logit[0];
#pragma unroll
        for (int c = 1; c < C_TYPES; ++c) m = fmaxf(m, logit[c]);
        float wexp[C_TYPES];
        float ssum = 0.0f;
#pragma unroll
        for (int c = 0; c < C_TYPES; ++c) { wexp[c] = __expf(logit[c] - m); ssum += wexp[c]; }
        const float inv = 1.0f / ssum;

#pragma unroll
        for (int c = 0; c < C_TYPES; ++c) {
            const float wc = wexp[c] * inv;
#pragma unroll
            for (int q = 0; q < 8; ++q) acc[c][q] += wc * votes[c][q];
        }
    };

    // Double-buffered TDM pipeline over this wave's capsules.
    prefetch(0, 0);
    for (int j = 0; j < CAPS_PER_WAVE - 1; ++j) {
        prefetch(j + 1, (j + 1) & 1);
        __builtin_amdgcn_s_wait_tensorcnt((short)2);  // slot j ready
        body(j);
    }
    __builtin_amdgcn_s_wait_tensorcnt((short)0);
    body(CAPS_PER_WAVE - 1);

    // All waves done with the TDM pool -> reuse it as the reduction stage.
    __syncthreads();
    float* stage = pool;                    // [wave][c][n][o]
#pragma unroll
    for (int c = 0; c < C_TYPES; ++c) {
        float* sp = &stage[wave * SVEC + c * 256 + n * 16 + 8 * half];
#pragma unroll
        for (int q = 0; q < 8; ++q) sp[q] = acc[c][q];
    }
    __syncthreads();

    float* pdst = partial + (size_t)(ch * N_BT + bt) * SVEC;
    for (int idx = tid; idx < SVEC; idx += 128) {
        pdst[idx] = stage[0 * SVEC + idx] + stage[1 * SVEC + idx] +
                    stage[2 * SVEC + idx] + stage[3 * SVEC + idx];
    }
}

// ---------------------------------------------------------------------------
// Sum chunk partials, add bias, squash; accumulate Vsum (t<2) or emit outputs.
// One thread per (b, c, o); 16-lane shuffle groups reduce over o.
// ---------------------------------------------------------------------------
__global__ __launch_bounds__(256) void caps_reduce_kernel(
    const float* __restrict__ partial,
    const float* __restrict__ bias,      // (10,16)
    float* __restrict__ Vsum,            // (256,10,16)
    float* __restrict__ out_poses,       // (256,10,16)
    float* __restrict__ out_acts,        // (256,10)
    int final_iter)
{
    const int g = blockIdx.x * 256 + threadIdx.x;   // 0 .. 40959
    const int o = g & 15;
    const int p = g >> 4;
    const int b = p / C_TYPES;
    const int c = p % C_TYPES;
    const int bt = b >> 4;
    const int n  = b & 15;

    float s = bias[c * POSE + o];
#pragma unroll
    for (int ch = 0; ch < NCHUNK; ++ch)
        s += partial[(size_t)(ch * N_BT + bt) * SVEC + c * 256 + n * 16 + o];

    float n2 = s * s;
    n2 += __shfl_xor(n2, 1, 16);
    n2 += __shfl_xor(n2, 2, 16);
    n2 += __shfl_xor(n2, 4, 16);
    n2 += __shfl_xor(n2, 8, 16);

    const float v = (n2 / (1.0f + n2)) * s * rsqrtf(n2 + CAPS_EPS);

    if (final_iter) {
        out_poses[(b * C_TYPES + c) * POSE + o] = v;
        float vn2 = v * v;
        vn2 += __shfl_xor(vn2, 1, 16);
        vn2 += __shfl_xor(vn2, 2, 16);
        vn2 += __shfl_xor(vn2, 4, 16);
        vn2 += __shfl_xor(vn2, 8, 16);
        if (o == 0) out_acts[b * C_TYPES + c] = sqrtf(vn2 + CAPS_EPS);
    } else {
        Vsum[(b * C_TYPES + c) * POSE + o] += v;   // logits are linear in v
    }
}

// ---------------------------------------------------------------------------
extern "C" void kernel_launch(void* const* d_in, const int* in_sizes, int n_in,
                              void* d_out, int out_size, void* d_ws, size_t ws_size,
                              hipStream_t stream) {
    const float* poses  = (const float*)d_in[0];   // (256,32,8,8,16,1)
    // d_in[1] = input_caps_activations: unused by the reference
    const float* weight = (const float*)d_in[2];   // (32,8,8,10,16,16)
    const float* bias   = (const float*)d_in[3];   // (10,1,1,16,1)

    float* out_poses = (float*)d_out;                       // 256*10*16
    float* out_acts  = (float*)d_out + BATCH * C_TYPES * POSE;

    float* Vsum    = (float*)d_ws;                          // 40960 floats
    float* partial = Vsum + BATCH * C_TYPES * POSE;         // 32*16*2560 floats

    caps_zero_kernel<<<(BATCH * C_TYPES * POSE + 255) / 256, 256, 0, stream>>>(
        Vsum, BATCH * C_TYPES * POSE);

    for (int t = 0; t < 3; ++t) {
        caps_pass_kernel<<<dim3(N_BT, NCHUNK), 128, 0, stream>>>(
            poses, weight, Vsum, partial);
        caps_reduce_kernel<<<(BATCH * C_TYPES * POSE + 255) / 256, 256, 0, stream>>>(
            partial, bias, Vsum, out_poses, out_acts, t == 2 ? 1 : 0);
    }
}